// MixtureLowRankRNN_20066087207574
// MI455X (gfx1250) — compile-verified
//
#include <hip/hip_runtime.h>

// Problem constants (match reference)
#define HIDDEN 1024
#define RANK 4
#define INPUT 16
#define BATCH 32
#define TSTEPS 1024
#define ALPHA 0.1f
#define BASE_SCALE 500.0f

typedef __attribute__((ext_vector_type(2))) float v2f;
typedef __attribute__((ext_vector_type(8))) float v8f;
typedef __attribute__((ext_vector_type(4))) unsigned int u32x4;
typedef __attribute__((ext_vector_type(8))) int i32x8;
typedef __attribute__((ext_vector_type(4))) int i32x4;

// ---------------------------------------------------------------------------
// Kernel 1: u[b,t,h] = sum_i x[b,t,i] * I_mat[h,i]
// [BT=32768, K=16] x [K=16, H=1024] f32 GEMM via V_WMMA_F32_16X16X4_F32
// (4 chained WMMAs per 16x16 tile, full f32 precision for the drive term).
// ---------------------------------------------------------------------------
__global__ __launch_bounds__(256) void u_gemm_wmma(
    const float* __restrict__ x,      // [BATCH*TSTEPS, INPUT]
    const float* __restrict__ I_mat,  // [HIDDEN, INPUT]
    float* __restrict__ u)            // [BATCH*TSTEPS, HIDDEN]
{
    const int wave = threadIdx.x >> 5;
    const int lane = threadIdx.x & 31;
    const int half = lane >> 4;      // 0: K pair {0,1}, 1: K pair {2,3}
    const int l16  = lane & 15;

    const long tile = (long)blockIdx.x * 8 + wave;       // 8 waves per block
    const int nTilesH = HIDDEN / 16;                     // 64
    const int mt = (int)(tile / nTilesH);                // bt-tile, 0..2047
    const int nt = (int)(tile % nTilesH);                // h-tile,  0..63

    const float* xrow = x     + (size_t)(mt * 16 + l16) * INPUT;  // A row M=l16
    const float* irow = I_mat + (size_t)(nt * 16 + l16) * INPUT;  // B col N=l16

    v8f acc = {};
#pragma unroll
    for (int k0 = 0; k0 < INPUT; k0 += 4) {
        v2f a, b;
        a.x = xrow[k0 + 2 * half + 0];
        a.y = xrow[k0 + 2 * half + 1];
        b.x = irow[k0 + 2 * half + 0];
        b.y = irow[k0 + 2 * half + 1];
        acc = __builtin_amdgcn_wmma_f32_16x16x4_f32(
            /*neg_a=*/false, a, /*neg_b=*/false, b,
            /*c_mod=*/(short)0, acc, /*reuse_a=*/false, /*reuse_b=*/false);
    }

    float* urow = u + (size_t)(mt * 16 + 8 * half) * HIDDEN + nt * 16 + l16;
#pragma unroll
    for (int v = 0; v < 8; ++v)
        urow[(size_t)v * HIDDEN] = acc[v];
}

// ---------------------------------------------------------------------------
// Kernel 2: sequential scan. One workgroup per batch, 256 threads, h in
// registers (4 f32/thread). Per step: tanh -> z (shfl_xor + 1 LDS barrier)
// -> h update -> 4-dim output s[0:4]. The u row for step t+1 is DMAed into
// LDS by the Tensor Data Mover (one tensor_load_to_lds per step, issued by
// wave 0, TENSORcnt-tracked, triple-buffered) while the VALU computes step t.
// ---------------------------------------------------------------------------
__global__ __launch_bounds__(256) void rnn_scan(
    const float* __restrict__ x,      // [BATCH, TSTEPS, INPUT]
    const float* __restrict__ m,      // [HIDDEN, RANK]
    const float* __restrict__ n,      // [HIDDEN, RANK]
    const float* __restrict__ I_mat,  // [HIDDEN, INPUT]
    const float* __restrict__ u,      // [BATCH, TSTEPS, HIDDEN] or null
    float* __restrict__ out,          // [BATCH, TSTEPS, 20]
    int use_u)
{
    const int b    = blockIdx.x;
    const int tid  = threadIdx.x;
    const int j0   = tid * 4;
    const int wave = tid >> 5;
    const int lane = tid & 31;

    const float alpha = ALPHA;
    const float oma   = 1.0f - ALPHA;
    const float scale = BASE_SCALE / (float)HIDDEN;

    __shared__ float zpart[2][8][4];     // double-buffered z partials
    __shared__ float xs[2][16];          // staged x_t (fallback path)
    __shared__ float ubuf[3][HIDDEN];    // TDM triple buffer (3 x 4 KB)

    // Preload low-rank factors for this thread's 4 hidden rows.
    float nreg[4][4], mreg[4][4];
#pragma unroll
    for (int jj = 0; jj < 4; ++jj)
#pragma unroll
        for (int r = 0; r < 4; ++r) {
            nreg[jj][r] = n[(j0 + jj) * RANK + r];
            mreg[jj][r] = m[(j0 + jj) * RANK + r];
        }

    float Ireg[4][16];
    if (!use_u) {
#pragma unroll
        for (int jj = 0; jj < 4; ++jj)
#pragma unroll
            for (int i = 0; i < 16; ++i)
                Ireg[jj][i] = I_mat[(j0 + jj) * INPUT + i];
    }

    const float* ubase = use_u ? (u + (size_t)b * TSTEPS * HIDDEN) : nullptr;
    const float* xb = x + (size_t)b * TSTEPS * INPUT;
    float* ob = out + (size_t)b * TSTEPS * 20;

    // LDS byte offsets of the triple buffer (flat aperture: LDS off = addr[31:0])
    unsigned int ldsOff[3];
#pragma unroll
    for (int q = 0; q < 3; ++q)
        ldsOff[q] = (unsigned int)(uintptr_t)(void*)&ubuf[q][0];

    // Issue one TDM load: u row (4 KB) for step t -> ubuf[t % 3].
    auto tdm_issue = [&](int t) {
        unsigned long long ga =
            (unsigned long long)(uintptr_t)(ubase + (size_t)t * HIDDEN);
        u32x4 g0;
        g0.x = 1u;                                   // count=1, user D#
        g0.y = ldsOff[t % 3];                        // lds_addr (bits 63:32)
        g0.z = (unsigned int)(ga & 0xffffffffull);   // global_addr lo
        g0.w = (unsigned int)((ga >> 32) & 0x01ffffffull) | (2u << 30); // hi + type=2
        i32x8 g1;
        g1[0] = (int)(2u << 16);               // data_size = 4 bytes
        g1[1] = (int)((unsigned)HIDDEN << 16); // tensor_dim0 = 1024 (lo16)
        g1[2] = (int)(1u << 16);               // tensor_dim0 hi = 0, tensor_dim1 = 1
        g1[3] = (int)((unsigned)HIDDEN << 16); // tile_dim0 = 1024
        g1[4] = 1;                             // tile_dim1 = 1, tile_dim2 = 0
        g1[5] = HIDDEN;                        // tensor_dim0_stride = 1024
        g1[6] = 0;
        g1[7] = 0;
        i32x4 gz4 = {0, 0, 0, 0};              // groups 2/3 unused (2-D tile)
        i32x8 gz8 = {0, 0, 0, 0, 0, 0, 0, 0};  // trailing group (clang-23 form)
        __builtin_amdgcn_tensor_load_to_lds(g0, g1, gz4, gz4, gz8, 0);
    };

    if (use_u && wave == 0) tdm_issue(0);   // prefetch row 0

    float hreg[4] = {0.f, 0.f, 0.f, 0.f};
    float sreg[4] = {0.f, 0.f, 0.f, 0.f};   // s[0:4] = (proj@h)[0:4]

    for (int t = 0; t < TSTEPS; ++t) {
        // Stage next step's drive term while this step computes.
        if (use_u) {
            if (wave == 0 && t + 1 < TSTEPS) tdm_issue(t + 1);
        } else {
            if (tid < 16) xs[t & 1][tid] = xb[(size_t)t * INPUT + tid];
        }

        // tanh + per-thread z partials
        float zp0 = 0.f, zp1 = 0.f, zp2 = 0.f, zp3 = 0.f;
#pragma unroll
        for (int jj = 0; jj < 4; ++jj) {
            float th = tanhf(hreg[jj]);
            zp0 += th * nreg[jj][0];
            zp1 += th * nreg[jj][1];
            zp2 += th * nreg[jj][2];
            zp3 += th * nreg[jj][3];
        }
        // wave32 butterfly reduction
#pragma unroll
        for (int off = 16; off > 0; off >>= 1) {
            zp0 += __shfl_xor(zp0, off, 32);
            zp1 += __shfl_xor(zp1, off, 32);
            zp2 += __shfl_xor(zp2, off, 32);
            zp3 += __shfl_xor(zp3, off, 32);
        }
        if (lane == 0) {
            zpart[t & 1][wave][0] = zp0;
            zpart[t & 1][wave][1] = zp1;
            zpart[t & 1][wave][2] = zp2;
            zpart[t & 1][wave][3] = zp3;
        }

        // Row t must be in LDS before anyone reads it after the barrier.
        // <=1 outstanding means the (older) row-t transfer has completed.
        if (use_u && wave == 0) {
            if (t + 1 < TSTEPS) __builtin_amdgcn_s_wait_tensorcnt(1);
            else                __builtin_amdgcn_s_wait_tensorcnt(0);
        }
        __syncthreads();  // single barrier per step (z partials + TDM row)

        float z[4];
#pragma unroll
        for (int r = 0; r < 4; ++r) {
            float a = 0.f;
#pragma unroll
            for (int w = 0; w < 8; ++w) a += zpart[t & 1][w][r];
            z[r] = a;
        }

        // drive term
        float uj[4];
        if (use_u) {
            const float4 q = *(const float4*)&ubuf[t % 3][j0];
            uj[0] = q.x; uj[1] = q.y; uj[2] = q.z; uj[3] = q.w;
        } else {
#pragma unroll
            for (int jj = 0; jj < 4; ++jj) {
                float a = 0.f;
#pragma unroll
                for (int i = 0; i < 16; ++i) a += Ireg[jj][i] * xs[t & 1][i];
                uj[jj] = a;
            }
        }

        // h update: h = (1-a)h + a*(scale * m@z + u)
#pragma unroll
        for (int jj = 0; jj < 4; ++jj) {
            float rec = z[0] * mreg[jj][0] + z[1] * mreg[jj][1] +
                        z[2] * mreg[jj][2] + z[3] * mreg[jj][3];
            hreg[jj] = oma * hreg[jj] + alpha * (scale * rec + uj[jj]);
        }

        // s[0:4] = (1-a)s + a*scale*z ; all threads hold identical z.
#pragma unroll
        for (int r = 0; r < 4; ++r)
            sreg[r] = oma * sreg[r] + (alpha * scale) * z[r];
        if (tid < 4) ob[(size_t)t * 20 + tid] = sreg[tid];
    }
}

// ---------------------------------------------------------------------------
// Kernel 3: out[b,t,4+i] = EWMA of x — rows 4..19 of proj@h need no hidden
// state (proj@span = I  =>  proj@I_mat = [0; I16]). 512 independent scans.
// ---------------------------------------------------------------------------
__global__ void x_ewma(const float* __restrict__ x, float* __restrict__ out)
{
    int g = blockIdx.x * blockDim.x + threadIdx.x;
    if (g >= BATCH * INPUT) return;
    int b = g >> 4;
    int i = g & 15;
    const float* xb = x + (size_t)b * TSTEPS * INPUT + i;
    float* ob = out + (size_t)b * TSTEPS * 20 + 4 + i;
    float s = 0.f;
    for (int t = 0; t < TSTEPS; ++t) {
        s = (1.0f - ALPHA) * s + ALPHA * xb[(size_t)t * INPUT];
        ob[(size_t)t * 20] = s;
    }
}

// ---------------------------------------------------------------------------
extern "C" void kernel_launch(void* const* d_in, const int* in_sizes, int n_in,
                              void* d_out, int out_size, void* d_ws, size_t ws_size,
                              hipStream_t stream)
{
    const float* x     = (const float*)d_in[0];  // [32,1024,16]
    const float* m     = (const float*)d_in[1];  // [1024,4]
    const float* n     = (const float*)d_in[2];  // [1024,4]
    const float* I_mat = (const float*)d_in[3];  // [1024,16]
    float* out = (float*)d_out;                  // [32,1024,20]

    const size_t uBytes = (size_t)BATCH * TSTEPS * HIDDEN * sizeof(float); // 128 MiB
    const int use_u = (d_ws != nullptr && ws_size >= uBytes) ? 1 : 0;
    float* u = (float*)d_ws;

    if (use_u) {
        // 2048 bt-tiles * 64 h-tiles = 131072 tiles; 8 waves/block
        const int totalTiles = (BATCH * TSTEPS / 16) * (HIDDEN / 16);
        u_gemm_wmma<<<totalTiles / 8, 256, 0, stream>>>(x, I_mat, u);
    }
    rnn_scan<<<BATCH, 256, 0, stream>>>(x, m, n, I_mat, use_u ? u : nullptr, out, use_u);
    x_ewma<<<(BATCH * INPUT + 255) / 256, 256, 0, stream>>>(x, out);
}